// GraphConvolution_83107617177960
// MI455X (gfx1250) — compile-verified
//
#include <hip/hip_runtime.h>
#include <hip/hip_bf16.h>

#define NN 100000
#define NE 1600000
#define DIN 128
#define DOUT 32

typedef __attribute__((ext_vector_type(2))) float v2f;
typedef __attribute__((ext_vector_type(8))) float v8f;

// ---------------------------------------------------------------------------
// 0) zero accumulation buffers (d_out is poisoned by the harness)
__global__ void init_kernel(float* __restrict__ out, float* __restrict__ deg) {
    int i = blockIdx.x * blockDim.x + threadIdx.x;
    int stride = gridDim.x * blockDim.x;
    for (int j = i; j < NN * DOUT; j += stride) out[j] = 0.0f;
    for (int j = i; j < NN; j += stride) deg[j] = 0.0f;
}

// ---------------------------------------------------------------------------
// 1) out-degree via scatter-add
__global__ void deg_kernel(const int* __restrict__ src, float* __restrict__ deg) {
    int e = blockIdx.x * blockDim.x + threadIdx.x;
    if (e < NE) atomicAdd(&deg[src[e]], 1.0f);
}

// 2) norm = rsqrt(max(deg,1)) in-place
__global__ void norm_kernel(float* __restrict__ deg) {
    int i = blockIdx.x * blockDim.x + threadIdx.x;
    if (i < NN) deg[i] = __frsqrt_rn(fmaxf(deg[i], 1.0f));
}

// ---------------------------------------------------------------------------
// 3) hn = (x @ W^T + b) * norm[row]   via V_WMMA_F32_16X16X4_F32
//    One wave (32 lanes) computes a 16x32 output tile. K=128 in steps of 4.
//    A (16x4 f32, 2 VGPRs): lanes 0-15 -> M=lane, K={0,1}; lanes 16-31 -> K={2,3}
//    B (4x16 f32, 2 VGPRs): lanes 0-15 -> N=lane, K={0,1}; lanes 16-31 -> K={2,3}
//    C/D (16x16 f32, 8 VGPRs): reg r, lanes 0-15 -> M=r; lanes 16-31 -> M=r+8
__global__ __launch_bounds__(256) void gemm_norm_kernel(
    const float* __restrict__ x, const float* __restrict__ W,
    const float* __restrict__ b, const float* __restrict__ norm,
    float* __restrict__ hn)
{
    const int ntiles = NN / 16;  // 6250
    int wave = blockIdx.x * (blockDim.x >> 5) + (threadIdx.x >> 5);
    if (wave >= ntiles) return;  // wave-uniform: EXEC stays all-ones for WMMA

    int lane = threadIdx.x & 31;
    int half = lane >> 4;        // which K/M half of the wave
    int l    = lane & 15;
    int row0 = wave * 16;

    // per-lane base pointers, pre-offset by the half-wave K shift (+2 elems)
    const float* xa  = x + (size_t)(row0 + l) * DIN + half * 2;   // A rows
    const float* wb0 = W + (size_t)l        * DIN + half * 2;     // B cols 0..15  (W rows)
    const float* wb1 = W + (size_t)(l + 16) * DIN + half * 2;     // B cols 16..31

    float bias0 = b[l];
    float bias1 = b[l + 16];
    v8f acc0, acc1;
#pragma unroll
    for (int r = 0; r < 8; ++r) { acc0[r] = bias0; acc1[r] = bias1; }

#pragma unroll
    for (int k0 = 0; k0 < DIN; k0 += 4) {
        v2f a, bv0, bv1;
        a.x   = xa[k0];   a.y   = xa[k0 + 1];
        bv0.x = wb0[k0];  bv0.y = wb0[k0 + 1];
        bv1.x = wb1[k0];  bv1.y = wb1[k0 + 1];
        acc0 = __builtin_amdgcn_wmma_f32_16x16x4_f32(
            false, a, false, bv0, (short)0, acc0, false, false);
        acc1 = __builtin_amdgcn_wmma_f32_16x16x4_f32(
            false, a, false, bv1, (short)0, acc1, false, false);
    }

#pragma unroll
    for (int r = 0; r < 8; ++r) {
        int m   = r + half * 8;
        int row = row0 + m;
        float s = norm[row];                     // fold norm[src-node] here
        hn[(size_t)row * DOUT + l]      = acc0[r] * s;
        hn[(size_t)row * DOUT + 16 + l] = acc1[r] * s;
    }
}

// ---------------------------------------------------------------------------
// 4) per-edge: out[dst] += hn[src] * norm[dst]; 8 lanes per edge, float4 each
__global__ __launch_bounds__(256) void edge_kernel(
    const float* __restrict__ hn, const float* __restrict__ norm,
    const int* __restrict__ src, const int* __restrict__ dst,
    float* __restrict__ out)
{
    long long t = (long long)blockIdx.x * blockDim.x + threadIdx.x;
    int e = (int)(t >> 3);
    if (e >= NE) return;
    int c = ((int)t & 7) * 4;

    int s = src[e];
    int d = dst[e];
    float nd = norm[d];

    const float4* hp = (const float4*)(hn + (size_t)s * DOUT + c);
    float4 v = *hp;                 // global_load_b128, L2-resident gather
    float* op = out + (size_t)d * DOUT + c;
    atomicAdd(op + 0, v.x * nd);
    atomicAdd(op + 1, v.y * nd);
    atomicAdd(op + 2, v.z * nd);
    atomicAdd(op + 3, v.w * nd);
}

// ---------------------------------------------------------------------------
// 5) relu in place on d_out
__global__ void relu_kernel(float* __restrict__ out) {
    int i = blockIdx.x * blockDim.x + threadIdx.x;
    if (i < NN * DOUT) out[i] = fmaxf(out[i], 0.0f);
}

// ---------------------------------------------------------------------------
extern "C" void kernel_launch(void* const* d_in, const int* in_sizes, int n_in,
                              void* d_out, int out_size, void* d_ws, size_t ws_size,
                              hipStream_t stream) {
    (void)in_sizes; (void)n_in; (void)out_size; (void)ws_size;
    const float* x  = (const float*)d_in[0];
    const float* W  = (const float*)d_in[1];
    const float* b  = (const float*)d_in[2];
    const int*  src = (const int*)d_in[3];
    const int*  dst = (const int*)d_in[4];
    float* out = (float*)d_out;

    float* hn  = (float*)d_ws;                       // [NN, 32]  12.8 MB
    float* deg = hn + (size_t)NN * DOUT;             // [NN]      0.4 MB (deg -> norm)

    init_kernel<<<2048, 256, 0, stream>>>(out, deg);
    deg_kernel<<<(NE + 255) / 256, 256, 0, stream>>>(src, deg);
    norm_kernel<<<(NN + 255) / 256, 256, 0, stream>>>(deg);

    const int ntiles = NN / 16;                      // 6250 waves
    const int wpb = 256 / 32;                        // 8 waves / block
    gemm_norm_kernel<<<(ntiles + wpb - 1) / wpb, 256, 0, stream>>>(x, W, b, deg, hn);

    long long ethreads = (long long)NE * 8;          // 12.8M
    edge_kernel<<<(int)((ethreads + 255) / 256), 256, 0, stream>>>(hn, deg, src, dst, out);

    relu_kernel<<<(NN * DOUT + 255) / 256, 256, 0, stream>>>(out);
}